// STFT_65360812311189
// MI455X (gfx1250) — compile-verified
//
#include <hip/hip_runtime.h>
#include <math.h>

#define NFFT   1024
#define HOP    256
#define NSAMP  524288
#define TILE_T 8                     // output frames per block
#define FRAMES (TILE_T + 1)          // +1 frame for phase-diff seed
#define SMPL   (TILE_T * HOP + NFFT) // 3072 staged samples
#define NBINS  512                   // k = 1..512 (DC dropped)
#define NTOUT  2048
#define PI_F     3.14159265358979323846f
#define TWOPI_F  6.28318530717958647692f

typedef unsigned int v4u __attribute__((ext_vector_type(4)));
typedef int          v8i __attribute__((ext_vector_type(8)));
typedef int          v4i __attribute__((ext_vector_type(4)));
typedef float        v2f __attribute__((ext_vector_type(2)));
typedef float        v8f __attribute__((ext_vector_type(8)));

struct cplx { float x, y; };
__device__ __forceinline__ cplx cadd(cplx a, cplx b){ return {a.x+b.x, a.y+b.y}; }
__device__ __forceinline__ cplx csub(cplx a, cplx b){ return {a.x-b.x, a.y-b.y}; }
__device__ __forceinline__ cplx cmul(cplx a, cplx b){ return {a.x*b.x - a.y*b.y, a.x*b.y + a.y*b.x}; }

__global__ __launch_bounds__(256)
void stft_fused_kernel(const float* __restrict__ x, float* __restrict__ out)
{
    __shared__ float s_x[SMPL];            // 12 KB staged samples
    __shared__ cplx  s_a[NFFT];            // 8 KB FFT ping
    __shared__ cplx  s_b[NFFT];            // 8 KB FFT pong / WMMA result (re|im)
    __shared__ float s_mag[NBINS*TILE_T];  // 16 KB output staging
    __shared__ float s_ph [NBINS*TILE_T];  // 16 KB output staging

    const int tid  = threadIdx.x;
    const int lane = tid & 31;
    const int wid  = tid >> 5;             // 8 waves
    const int tile = blockIdx.x & 255;     // 256 tiles of 8 frames -> 2048 out cols
    const int b    = blockIdx.x >> 8;      // batch
    const int t0   = tile * TILE_T;        // first frame computed (seed frame)
    const long s0  = (long)t0 * HOP - 512; // x-index of first staged sample

    // ---------------- stage input window into LDS (TDM for interior tiles) ----------------
    const bool interior = (tile >= 1) && (tile <= 254);
    if (interior) {
        if (tid < 32) { // one wave issues the TDM descriptor
            const float* src = x + (size_t)b * NSAMP + s0;
            unsigned long long ga = (unsigned long long)(uintptr_t)src;
            unsigned int lds = (unsigned int)(uintptr_t)(void*)&s_x[0];
            v4u g0 = { 1u,
                       lds,
                       (unsigned int)(ga & 0xFFFFFFFFu),
                       (unsigned int)(((ga >> 32) & 0x01FFFFFFu) | (2u << 30)) };
            v8i g1 = { (int)0x00020000u,                  // data_size = 4B
                       (int)((SMPL & 0xFFFF) << 16),      // tensor_dim0 lo16
                       (int)(((SMPL >> 16) & 0xFFFF) | (1u << 16)), // dim0 hi16 | tensor_dim1=1
                       (int)((SMPL & 0xFFFF) << 16),      // tile_dim0
                       0, (int)SMPL, 0, 0 };              // tensor_dim0_stride
            v4i gz4 = {0, 0, 0, 0};
            v8i gz8 = {0, 0, 0, 0, 0, 0, 0, 0};
            __builtin_amdgcn_tensor_load_to_lds(g0, g1, gz4, gz4, gz8, 0);
            __builtin_amdgcn_s_wait_tensorcnt(0);
        }
    } else {
        for (int i = tid; i < SMPL; i += 256) {
            long j = s0 + i;
            if (j < 0) j = -j;
            if (j >= NSAMP) j = 2L*(NSAMP-1) - j;
            s_x[i] = x[(size_t)b * NSAMP + j];
        }
    }

    // ---------------- per-wave F16 DFT coefficient chunks (A-matrix, 16x4 f32 layout) ----------------
    // A layout: lanes 0-15 hold (M=lane, K=0) in elem0, K=1 in elem1; lanes 16-31 hold K=2, K=3.
    // F16[k1,n1] = cos(th) - i sin(th), th = 2*pi*k1*n1/16
    const int k1l   = lane & 15;
    const int nbase = (lane < 16) ? 0 : 2;
    v2f cosA[4], sinA[4], nsinA[4];
#pragma unroll
    for (int c4 = 0; c4 < 4; ++c4) {
        int n1a = 4*c4 + nbase;
        float tha = (TWOPI_F/16.0f) * (float)(k1l * n1a);
        float thb = (TWOPI_F/16.0f) * (float)(k1l * (n1a+1));
        float sa, ca, sb, cb;
        sincosf(tha, &sa, &ca);
        sincosf(thb, &sb, &cb);
        cosA[c4]  = { ca,  cb };
        sinA[c4]  = { sa,  sb };
        nsinA[c4] = {-sa, -sb };
    }
    const int nt   = wid & 3;   // N-tile (16 of the 64 k2 columns)
    const int comp = wid >> 2;  // 0 -> real part of Z, 1 -> imag part
    __syncthreads();

    float* resr = (float*)s_b;        // 1024 floats: Re X[k]
    float* resi = resr + NFFT;        // 1024 floats: Im X[k]

    float pp[2] = {0.f, 0.f};

    for (int f = 0; f < FRAMES; ++f) {
        // ---- gather + window: z[n1*64 + n2] = win(n)*x[n], n = n1 + 16*n2 ----
#pragma unroll
        for (int j = 0; j < 4; ++j) {
            int i  = tid + 256*j;
            int n1 = i >> 6, n2 = i & 63;
            int n  = n1 + 16*n2;
            float w = 0.5f * (1.0f - cosf(TWOPI_F * (float)n * (1.0f/1024.0f)));
            s_a[i] = { s_x[f*HOP + n] * w, 0.f };
        }
        __syncthreads();

        // ---- 16 independent 64-pt Stockham FFTs over n2 (6 radix-2 stages) ----
        cplx* pin  = s_a;
        cplx* pout = s_b;
        int m = 32, s = 1;
#pragma unroll
        for (int st = 0; st < 6; ++st) {
#pragma unroll
            for (int r = 0; r < 2; ++r) {
                int w = tid + 256*r;       // 0..511 = 16 groups x 32 butterflies
                int g = w >> 5;            // sub-FFT id (n1)
                int u = w & 31;
                int p = u / s;             // s is unrolled constant
                int q = u - p*s;
                float ang = -(float)(p*s) * (TWOPI_F / 64.0f);
                float sw, cw;
                sincosf(ang, &sw, &cw);
                cplx a  = pin[g*64 + q + s*p];
                cplx bb = pin[g*64 + q + s*(p+m)];
                pout[g*64 + q + s*(2*p)]     = cadd(a, bb);
                cplx tw = {cw, sw};
                pout[g*64 + q + s*(2*p + 1)] = cmul(csub(a, bb), tw);
            }
            __syncthreads();
            cplx* t = pin; pin = pout; pout = t;
            m >>= 1; s <<= 1;
        }
        // 6 stages -> result (natural order per group) back in s_a; s_b free

        // ---- twiddle: Y[n1,k2] = W1024^(n1*k2) * A64[n1,k2], in place in s_a ----
#pragma unroll
        for (int j = 0; j < 4; ++j) {
            int i  = tid + 256*j;
            int n1 = i >> 6, k2 = i & 63;
            float ang = -(float)(n1 * k2) * (TWOPI_F / 1024.0f);
            float sw, cw;
            sincosf(ang, &sw, &cw);
            cplx tw = {cw, sw};
            s_a[i] = cmul(s_a[i], tw);
        }
        __syncthreads();

        // ---- final 16-pt DFT stage as WMMA: Z(16x64) = F16(16x16) x Y(16x64), complex ----
        // wave (nt, comp): 16x16 f32 tile via 8 chained V_WMMA_F32_16X16X4_F32
        {
            const int k2 = nt*16 + (lane & 15);
            v2f Br[4], Bi[4];
#pragma unroll
            for (int c4 = 0; c4 < 4; ++c4) {
                int n1a = 4*c4 + nbase;          // B 4x16: lanes 0-15 rows K=0/1, lanes 16-31 rows K=2/3
                cplx ya = s_a[n1a*64 + k2];
                cplx yb = s_a[(n1a+1)*64 + k2];
                Br[c4] = { ya.x, yb.x };
                Bi[c4] = { ya.y, yb.y };
            }
            v8f acc = {0.f,0.f,0.f,0.f,0.f,0.f,0.f,0.f};
            // Zr = sum cos*Yr + sin*Yi ; Zi = sum cos*Yi - sin*Yr
#pragma unroll
            for (int c4 = 0; c4 < 4; ++c4) {
                v2f Bv = (comp == 0) ? Br[c4] : Bi[c4];
                acc = __builtin_amdgcn_wmma_f32_16x16x4_f32(
                        false, cosA[c4], false, Bv, (short)0, acc, false, false);
            }
#pragma unroll
            for (int c4 = 0; c4 < 4; ++c4) {
                v2f Av = (comp == 0) ? sinA[c4] : nsinA[c4];
                v2f Bv = (comp == 0) ? Bi[c4]   : Br[c4];
                acc = __builtin_amdgcn_wmma_f32_16x16x4_f32(
                        false, Av, false, Bv, (short)0, acc, false, false);
            }
            // D layout: VGPR v holds M=v (lanes 0-15) / M=v+8 (lanes 16-31); X[k2 + 64*k1]
            float* dst = comp ? resi : resr;
            const int mbase = (lane < 16) ? 0 : 8;
#pragma unroll
            for (int v = 0; v < 8; ++v)
                dst[k2 + 64*(v + mbase)] = acc[v];
        }
        __syncthreads();

        // ---- mag / phase post-process (pointwise diff-of-unwrap) ----
#pragma unroll
        for (int r = 0; r < 2; ++r) {
            int kk = tid + 256*r;          // bin k = kk+1
            float re = resr[kk + 1];
            float im = resi[kk + 1];
            float mag = sqrtf(re*re + im*im);
            float lm  = (logf(fmaxf(mag, 1e-7f)) + 17.0f) * (1.0f/23.0f);
            float ph  = atan2f(im, re);
            if (f > 0) {
                float dd = ph - pp[r];
                float md = fmodf(dd + PI_F, TWOPI_F);
                if (md < 0.f) md += TWOPI_F;
                float ddm = md - PI_F;
                if (ddm == -PI_F && dd > 0.f) ddm = PI_F;
                float dout = (fabsf(dd) < PI_F) ? dd : ddm;
                s_mag[kk*TILE_T + (f-1)] = lm;
                s_ph [kk*TILE_T + (f-1)] = dout;
            }
            pp[r] = ph;
        }
        __syncthreads();  // protect s_a/s_b reuse next frame
    }

    // ---------------- coalesced output: rows of 8 floats (32B, aligned) ----------------
    float* outm = out + ((size_t)(b*2 + 0) * NBINS) * NTOUT + t0;
    float* outp = out + ((size_t)(b*2 + 1) * NBINS) * NTOUT + t0;
#pragma unroll
    for (int r = 0; r < 2; ++r) {
        int kk = tid + 256*r;
        const float4* srcm = (const float4*)&s_mag[kk*TILE_T];
        const float4* srcp = (const float4*)&s_ph [kk*TILE_T];
        float4* dm = (float4*)(outm + (size_t)kk * NTOUT);
        float4* dp = (float4*)(outp + (size_t)kk * NTOUT);
        dm[0] = srcm[0]; dm[1] = srcm[1];
        dp[0] = srcp[0]; dp[1] = srcp[1];
    }
}

extern "C" void kernel_launch(void* const* d_in, const int* in_sizes, int n_in,
                              void* d_out, int out_size, void* d_ws, size_t ws_size,
                              hipStream_t stream)
{
    const float* x = (const float*)d_in[0];
    float* out = (float*)d_out;
    stft_fused_kernel<<<dim3(32 * 256), dim3(256), 0, stream>>>(x, out);
}